// _GraphLaplacianDiffusion_33809982554551
// MI455X (gfx1250) — compile-verified
//
#include <hip/hip_runtime.h>
#include <stdint.h>

#define D_FEAT   64
#define CHUNK    256      // edges staged per block iteration
#define BLOCK    256      // 8 wave32 per block

// ===========================================================================
// CSR path (preferred): count -> scan -> fill(perm) -> sort rows -> gather-sum
// ===========================================================================

// --- zero the count/cursor buffer -----------------------------------------
__global__ void gld_zero_u32_kernel(uint32_t* __restrict__ p, int n) {
    const int stride = gridDim.x * blockDim.x;
    for (int k = blockIdx.x * blockDim.x + threadIdx.x; k < n; k += stride)
        p[k] = 0u;
}

// --- per-edge in-degree histogram (integer atomics) -----------------------
__global__ void gld_count_kernel(const int* __restrict__ dstRow,
                                 uint32_t* __restrict__ count, int n_edges) {
    const int stride = gridDim.x * blockDim.x;
    for (int e = blockIdx.x * blockDim.x + threadIdx.x; e < n_edges; e += stride)
        atomicAdd(&count[dstRow[e]], 1u);
}

// --- single-block exclusive scan: count -> rowStart; also re-zeroes count --
__global__ __launch_bounds__(256)
void gld_scan_kernel(uint32_t* __restrict__ count,
                     uint32_t* __restrict__ rowStart, int n) {
    __shared__ uint32_t ssum[256];
    const int t   = threadIdx.x;
    const int per = (n + 255) / 256;
    const int beg = t * per;
    const int end = (beg + per < n) ? (beg + per) : n;

    uint32_t local = 0;
    for (int i = beg; i < end; ++i) local += count[i];
    ssum[t] = local;
    __syncthreads();

    // Hillis-Steele inclusive scan over 256 partials
    for (int off = 1; off < 256; off <<= 1) {
        uint32_t v   = ssum[t];
        uint32_t add = (t >= off) ? ssum[t - off] : 0u;
        __syncthreads();
        ssum[t] = v + add;
        __syncthreads();
    }

    uint32_t off = (t == 0) ? 0u : ssum[t - 1];   // exclusive prefix of strip
    const uint32_t total = ssum[255];
    for (int i = beg; i < end; ++i) {
        uint32_t c = count[i];
        rowStart[i] = off;
        off += c;
        count[i] = 0u;                             // becomes fill cursor
    }
    if (t == 255) rowStart[n] = total;
}

// --- fill perm[] with src ids, CSR by dst; edge indices staged via async DMA
__global__ __launch_bounds__(BLOCK)
void gld_fill_kernel(const int* __restrict__ srcRow,
                     const int* __restrict__ dstRow,
                     const uint32_t* __restrict__ rowStart,
                     uint32_t* __restrict__ cursor,
                     uint32_t* __restrict__ perm,
                     int n_edges, int n_chunks) {
    __shared__ __align__(16) int s_src[CHUNK];
    __shared__ __align__(16) int s_dst[CHUNK];
    const int t = threadIdx.x;

    for (int c = blockIdx.x; c < n_chunks; c += gridDim.x) {
        const int base      = c * CHUNK;
        const int remaining = n_edges - base;

        if (remaining >= CHUNK) {
            // CDNA5 async global->LDS DMA: 128 lanes x 16B = 2 KB of indices.
            if (t < 128) {
                const int  q  = t & 63;
                const int* gp = ((t < 64) ? srcRow : dstRow) + base + q * 4;
                int*       lp = ((t < 64) ? s_src : s_dst) + q * 4;
                uint32_t ldsoff = (uint32_t)(uintptr_t)lp;   // low 32b == LDS offset
                uint64_t ga     = (uint64_t)(uintptr_t)gp;
                asm volatile("global_load_async_to_lds_b128 %0, %1, off"
                             :: "v"(ldsoff), "v"(ga) : "memory");
            }
            asm volatile("s_wait_asynccnt 0" ::: "memory");
        } else {
            if (t < remaining) {
                s_src[t] = srcRow[base + t];
                s_dst[t] = dstRow[base + t];
            }
        }
        __syncthreads();

        const int cnt = remaining < CHUNK ? remaining : CHUNK;
        if (t < cnt) {
            const int s = s_src[t];
            const int d = s_dst[t];
            const uint32_t pos = atomicAdd(&cursor[d], 1u);
            perm[rowStart[d] + pos] = (uint32_t)s;
        }
        __syncthreads();
    }
}

// --- per-row insertion sort of perm => bitwise-deterministic summation -----
__global__ void gld_sortrows_kernel(const uint32_t* __restrict__ rowStart,
                                    uint32_t* __restrict__ perm, int n_nodes) {
    const int stride = gridDim.x * blockDim.x;
    for (int v = blockIdx.x * blockDim.x + threadIdx.x; v < n_nodes; v += stride) {
        const uint32_t beg = rowStart[v], end = rowStart[v + 1];
        for (uint32_t i = beg + 1; i < end; ++i) {
            uint32_t key = perm[i];
            uint32_t j = i;
            while (j > beg && perm[j - 1] > key) { perm[j] = perm[j - 1]; --j; }
            perm[j] = key;
        }
    }
}

// --- one wave32 per node: atomic-free gather-sum, fused finalize -----------
__global__ __launch_bounds__(BLOCK)
void gld_gather_sum_kernel(const float* __restrict__ H,
                           const uint32_t* __restrict__ rowStart,
                           const uint32_t* __restrict__ perm,
                           float* __restrict__ out, int n_nodes) {
    const int warp = threadIdx.x >> 5;
    const int lane = threadIdx.x & 31;
    const int wstr = gridDim.x * (BLOCK / 32);

    for (int v = blockIdx.x * (BLOCK / 32) + warp; v < n_nodes; v += wstr) {
        const uint32_t beg = rowStart[v], end = rowStart[v + 1];
        float ax0 = 0.f, ay0 = 0.f, ax1 = 0.f, ay1 = 0.f;
        uint32_t i = beg;
        for (; i + 2 <= end; i += 2) {                 // 2-way MLP on gathers
            const uint32_t s0 = perm[i], s1 = perm[i + 1];
            const float2 h0 = *(const float2*)(H + (size_t)s0 * D_FEAT + lane * 2);
            const float2 h1 = *(const float2*)(H + (size_t)s1 * D_FEAT + lane * 2);
            ax0 += h0.x; ay0 += h0.y;
            ax1 += h1.x; ay1 += h1.y;
        }
        if (i < end) {
            const uint32_t s0 = perm[i];
            const float2 h0 = *(const float2*)(H + (size_t)s0 * D_FEAT + lane * 2);
            ax0 += h0.x; ay0 += h0.y;
        }
        const float ax = ax0 + ax1, ay = ay0 + ay1;
        float dg = (float)(end - beg);
        if (dg < 1.f) dg = 1.f;
        const float2 hv = *(const float2*)(H + (size_t)v * D_FEAT + lane * 2);
        float2 o;
        o.x = hv.x - ax / dg;
        o.y = hv.y - ay / dg;
        *(float2*)(out + (size_t)v * D_FEAT + lane * 2) = o;
    }
}

// ===========================================================================
// Fallback path (tiny workspace): direct f32 atomic scatter (round-1 design)
// ===========================================================================
__global__ void gld_init_kernel(float* __restrict__ out,
                                float* __restrict__ deg, int n_nodes) {
    const int stride = gridDim.x * blockDim.x;
    const int i = blockIdx.x * blockDim.x + threadIdx.x;
    const int total = n_nodes * D_FEAT;
    for (int k = i; k < total; k += stride) out[k] = 0.0f;
    for (int k = i; k < n_nodes; k += stride) deg[k] = 0.0f;
}

__global__ __launch_bounds__(BLOCK)
void gld_scatter_kernel(const float* __restrict__ H,
                        const int* __restrict__ srcRow,
                        const int* __restrict__ dstRow,
                        float* __restrict__ out,
                        float* __restrict__ deg,
                        int n_edges, int n_chunks) {
    __shared__ __align__(16) int s_src[CHUNK];
    __shared__ __align__(16) int s_dst[CHUNK];
    const int t = threadIdx.x, warp = t >> 5, lane = t & 31;

    for (int c = blockIdx.x; c < n_chunks; c += gridDim.x) {
        const int base = c * CHUNK;
        const int remaining = n_edges - base;
        if (remaining >= CHUNK) {
            if (t < 128) {
                const int  q  = t & 63;
                const int* gp = ((t < 64) ? srcRow : dstRow) + base + q * 4;
                int*       lp = ((t < 64) ? s_src : s_dst) + q * 4;
                uint32_t ldsoff = (uint32_t)(uintptr_t)lp;
                uint64_t ga     = (uint64_t)(uintptr_t)gp;
                asm volatile("global_load_async_to_lds_b128 %0, %1, off"
                             :: "v"(ldsoff), "v"(ga) : "memory");
            }
            asm volatile("s_wait_asynccnt 0" ::: "memory");
        } else if (t < remaining) {
            s_src[t] = srcRow[base + t];
            s_dst[t] = dstRow[base + t];
        }
        __syncthreads();

        const int cnt  = remaining < CHUNK ? remaining : CHUNK;
        const int ebeg = warp * 32;
        const int eend = (ebeg + 32 < cnt) ? (ebeg + 32) : cnt;
        for (int e = ebeg; e < eend; ++e) {
            const int s = s_src[e], d = s_dst[e];
            const float2 h = *(const float2*)(H + (size_t)s * D_FEAT + lane * 2);
            float* o = out + (size_t)d * D_FEAT + lane * 2;
            atomicAdd(o, h.x);
            atomicAdd(o + 1, h.y);
            if (lane == 0) atomicAdd(deg + d, 1.0f);
        }
        __syncthreads();
    }
}

__global__ void gld_finalize_kernel(const float* __restrict__ H,
                                    float* __restrict__ out,
                                    const float* __restrict__ deg, int n_nodes) {
    const int stride = gridDim.x * blockDim.x;
    const int total = n_nodes * D_FEAT;
    for (int k = blockIdx.x * blockDim.x + threadIdx.x; k < total; k += stride) {
        float dg = deg[k >> 6];
        dg = dg < 1.0f ? 1.0f : dg;
        out[k] = H[k] - out[k] / dg;
    }
}

// ===========================================================================
extern "C" void kernel_launch(void* const* d_in, const int* in_sizes, int n_in,
                              void* d_out, int out_size, void* d_ws, size_t ws_size,
                              hipStream_t stream) {
    const float* H  = (const float*)d_in[0];
    const int*   ei = (const int*)d_in[1];

    const int n_nodes = in_sizes[0] / D_FEAT;
    const int n_edges = in_sizes[1] / 2;
    const int* srcRow = ei;
    const int* dstRow = ei + n_edges;
    float* out = (float*)d_out;

    const int n_chunks = (n_edges + CHUNK - 1) / CHUNK;
    int sblocks = n_chunks < 1024 ? n_chunks : 1024;
    if (sblocks < 1) sblocks = 1;

    // Workspace layout (CSR path): cursor/count | rowStart | perm
    const size_t need = (size_t)n_nodes * 4 + (size_t)(n_nodes + 1) * 4 +
                        (size_t)n_edges * 4;

    if (ws_size >= need) {
        uint32_t* cursor   = (uint32_t*)d_ws;               // n_nodes
        uint32_t* rowStart = cursor + n_nodes;              // n_nodes + 1
        uint32_t* perm     = rowStart + (n_nodes + 1);      // n_edges

        gld_zero_u32_kernel<<<256, 256, 0, stream>>>(cursor, n_nodes);
        gld_count_kernel<<<1024, 256, 0, stream>>>(dstRow, cursor, n_edges);
        gld_scan_kernel<<<1, 256, 0, stream>>>(cursor, rowStart, n_nodes);
        gld_fill_kernel<<<sblocks, BLOCK, 0, stream>>>(srcRow, dstRow, rowStart,
                                                       cursor, perm,
                                                       n_edges, n_chunks);
        gld_sortrows_kernel<<<256, 256, 0, stream>>>(rowStart, perm, n_nodes);

        int gblocks = (n_nodes + (BLOCK / 32) - 1) / (BLOCK / 32);
        gld_gather_sum_kernel<<<gblocks, BLOCK, 0, stream>>>(H, rowStart, perm,
                                                             out, n_nodes);
    } else {
        float* deg = (float*)d_ws;                          // n_nodes floats
        gld_init_kernel<<<2048, 256, 0, stream>>>(out, deg, n_nodes);
        gld_scatter_kernel<<<sblocks, BLOCK, 0, stream>>>(H, srcRow, dstRow, out,
                                                          deg, n_edges, n_chunks);
        gld_finalize_kernel<<<2048, 256, 0, stream>>>(H, out, deg, n_nodes);
    }
}